// CycleGenerator_4063039062110
// MI455X (gfx1250) — compile-verified
//
#include <hip/hip_runtime.h>

#define N_NODES 55296
#define N_EDGES (4 * N_NODES)
#define F_IN    64
#define F_HID   256
#define BATCH   8
#define N_ROWS  (BATCH * N_NODES)   // 442368

typedef __attribute__((ext_vector_type(2))) float v2f;
typedef __attribute__((ext_vector_type(4))) float v4f;
typedef __attribute__((ext_vector_type(8))) float v8f;

typedef __attribute__((ext_vector_type(4))) unsigned int u32x4;
typedef __attribute__((ext_vector_type(8))) int          i32x8;
typedef __attribute__((ext_vector_type(4))) int          i32x4;

// Zero the sums region: out[row][0:64] for all rows (scatter target).
__global__ void zero_sums_kernel(float* __restrict__ out) {
    int idx = blockIdx.x * blockDim.x + threadIdx.x;   // N_ROWS*16 threads
    int row = idx >> 4;
    int q   = idx & 15;
    v4f z = {0.f, 0.f, 0.f, 0.f};
    *(v4f*)(out + (size_t)row * F_HID + q * 4) = z;
}

__global__ void zero_deg_kernel(float* __restrict__ deg) {
    int idx = blockIdx.x * blockDim.x + threadIdx.x;
    if (idx * 4 < N_NODES) {
        v4f z = {0.f, 0.f, 0.f, 0.f};
        *(v4f*)(deg + idx * 4) = z;
    }
}

__global__ void deg_kernel(const int* __restrict__ edge_dst, float* __restrict__ deg) {
    int e = blockIdx.x * blockDim.x + threadIdx.x;
    if (e < N_EDGES) atomicAdd(&deg[edge_dst[e]], 1.0f);
}

// One block = 4 edges x 64 features, grid.y = batch.
// Accumulates x[b, src, f] into out[b, dst, f] (f < 64); f32 atomics resolve in L2
// (the 113 MB sums slab fits in the 192 MB L2).
__global__ void scatter_kernel(const float* __restrict__ x,
                               const int* __restrict__ edge_src,
                               const int* __restrict__ edge_dst,
                               float* __restrict__ out) {
    int b = blockIdx.y;
    int e = blockIdx.x * 4 + (threadIdx.x >> 6);
    int f = threadIdx.x & 63;
    int src = edge_src[e];
    int dst = edge_dst[e];
    float v = x[((size_t)b * N_NODES + src) * F_IN + f];
    atomicAdd(out + ((size_t)b * N_NODES + dst) * F_HID + f, v);
}

// TDM: DMA `nelem` contiguous f32 from global into LDS at byte offset lds_off.
// D# per CDNA5 ISA ch.8: group0 = {count=1 | lds_addr | global_addr | type=2},
// group1 = {data_size=4B, tensor_dim0=nelem, tensor_dim1=1, tile_dim0=nelem,
//           tile_dim1=1, tensor_dim0_stride=nelem}, groups 2/3 unused (2D tensor).
// This toolchain's builtin takes 6 args (extra int32x8 before cpol).
__device__ __forceinline__ void tdm_load_f32(const float* gptr, unsigned lds_off,
                                             unsigned nelem) {
    unsigned long long ga = (unsigned long long)(uintptr_t)gptr;
    u32x4 g0 = { 1u,                                   // count=1, is_restore=0, gather=0
                 lds_off,                              // lds_addr (bytes)
                 (unsigned)(ga & 0xFFFFFFFFu),         // global_addr[31:0]
                 (unsigned)((ga >> 32) & 0x1FFFFFFu) | (2u << 30) }; // addr[56:32] | type=2
    i32x8 g1 = { (int)0x00020000,                // workgroup_mask=0, data_size=2 (4B)
                 (int)((nelem & 0xFFFFu) << 16), // tensor_dim0[15:0] in bits[63:48]
                 (int)0x00010000,                // tensor_dim0[31:16]=0, tensor_dim1=1
                 (int)((nelem & 0xFFFFu) << 16), // tile_dim0 in bits[127:112]
                 1,                              // tile_dim1=1, tile_dim2=0
                 (int)nelem,                     // tensor_dim0_stride[31:0]
                 0, 0 };                         // stride hi / dim1_stride = 0
    i32x4 g2 = {0, 0, 0, 0};
    i32x4 g3 = {0, 0, 0, 0};
    i32x8 g4 = {0, 0, 0, 0, 0, 0, 0, 0};
    __builtin_amdgcn_tensor_load_to_lds(g0, g1, g2, g3, g4, 0);
}

// One wave32 computes a 16-row x 256-col output slab using V_WMMA_F32_16X16X4_F32.
// Block = 256 threads = 8 waves = 128 rows. Grid = N_ROWS/128 blocks.
// Weights + bias staged in LDS once per block via the Tensor Data Mover;
// B-fragments and bias then come from the DS path (pipelined DScnt waits).
__global__ void __launch_bounds__(256)
gemm_kernel(const float* __restrict__ x,
            const float* __restrict__ W_l,
            const float* __restrict__ b_l,
            const float* __restrict__ W_r,
            const float* __restrict__ deg,
            float* __restrict__ out) {
    // 132096 B: [0,16384) = W_l, [16384,32768) = W_r, [32768,33024) = b_l
    extern __shared__ float smem[];

    if (threadIdx.x < 32) {           // wave 0 issues the DMAs (TDM ignores EXEC)
        tdm_load_f32(W_l, 0u,      16384u);
        tdm_load_f32(W_r, 65536u,  16384u);
        tdm_load_f32(b_l, 131072u, 256u);
        __builtin_amdgcn_s_wait_tensorcnt(0);
    }
    __syncthreads();

    const int lane = threadIdx.x & 31;
    const int wv   = threadIdx.x >> 5;
    const int rl   = lane & 15;       // row (A) / col (B,C) index within tile
    const int kh   = lane >> 4;       // K half: lanes 16..31 hold K offset +2
    const int rowbase = (blockIdx.x * 8 + wv) * 16;
    const int row     = rowbase + rl;

    const float d   = deg[row % N_NODES];
    const float inv = 1.0f / fmaxf(d, 1.0f);

    // A fragments: lane l<16 -> row rl, K = 4k+{0,1}; lane l>=16 -> K = 4k+2+{0,1}.
    v2f a1[16];  // agg = sums * (1/deg); sums live in out[row][0:64]
    v2f a2[16];  // x
    const float* sumsp = out + (size_t)row * F_HID + kh * 2;
    const float* xp    = x   + (size_t)row * F_IN  + kh * 2;
#pragma unroll
    for (int k = 0; k < 16; ++k) {
        v2f s  = *(const v2f*)(sumsp + k * 4);
        a1[k] = s * inv;
        a2[k] = *(const v2f*)(xp + k * 4);
    }

    for (int n = 0; n < 16; ++n) {
        const int col = n * 16 + rl;
        const float bias = smem[32768 + col];          // ds_load_b32
        v8f c = {bias, bias, bias, bias, bias, bias, bias, bias};

        // B fragment: lane l holds col rl of B = W^T, K pair 4k+2*kh+{0,1}
        // = W[col][4k+2*kh .. +1], a contiguous float2 -> DS loads from LDS.
        const float* wl = smem + col * F_IN + kh * 2;
        const float* wr = smem + 16384 + col * F_IN + kh * 2;
#pragma unroll
        for (int k = 0; k < 16; ++k) {
            v2f bf = *(const v2f*)(wl + k * 4);
            c = __builtin_amdgcn_wmma_f32_16x16x4_f32(
                    false, a1[k], false, bf, (short)0, c, false, false);
        }
#pragma unroll
        for (int k = 0; k < 16; ++k) {
            v2f bf = *(const v2f*)(wr + k * 4);
            c = __builtin_amdgcn_wmma_f32_16x16x4_f32(
                    false, a2[k], false, bf, (short)0, c, false, false);
        }

        // C layout: lane l, vgpr j -> M = j + 8*kh, N = rl.
        float* op = out + (size_t)(rowbase + kh * 8) * F_HID + col;
#pragma unroll
        for (int j = 0; j < 8; ++j) {
            op[(size_t)j * F_HID] = fmaxf(c[j], 0.0f);
        }
    }
}

extern "C" void kernel_launch(void* const* d_in, const int* in_sizes, int n_in,
                              void* d_out, int out_size, void* d_ws, size_t ws_size,
                              hipStream_t stream) {
    const float* x    = (const float*)d_in[0];
    const int*   esrc = (const int*)  d_in[1];
    const int*   edst = (const int*)  d_in[2];
    const float* W_l  = (const float*)d_in[3];
    const float* b_l  = (const float*)d_in[4];
    const float* W_r  = (const float*)d_in[5];
    float* out = (float*)d_out;
    float* deg = (float*)d_ws;   // 55296 floats

    zero_sums_kernel<<<(N_ROWS * 16) / 256, 256, 0, stream>>>(out);
    zero_deg_kernel<<<(N_NODES / 4 + 255) / 256, 256, 0, stream>>>(deg);
    deg_kernel<<<(N_EDGES + 255) / 256, 256, 0, stream>>>(edst, deg);

    dim3 sgrid(N_EDGES / 4, BATCH);
    scatter_kernel<<<sgrid, 256, 0, stream>>>(x, esrc, edst, out);

    gemm_kernel<<<N_ROWS / 128, 256, 132096, stream>>>(x, W_l, b_l, W_r, deg, out);
}